// SharedDeformConv2d_11553462026730
// MI455X (gfx1250) — compile-verified
//
#include <hip/hip_runtime.h>
#include <hip/hip_bf16.h>

typedef __attribute__((ext_vector_type(2))) float v2f;
typedef __attribute__((ext_vector_type(4))) float v4f;
typedef __attribute__((ext_vector_type(8))) float v8f;

#define B_    4
#define C_    256
#define H_    64
#define W_    64
#define O_    256
#define PIX   (H_ * W_)      // 4096
#define K2    9
#define KDIM  (C_ * K2)      // 2304
#define PAD_  2
#define DIL_  2

#define TM       64          // M-tile (pixels) per workgroup (4 WMMA tiles/wave)
#define KC       64          // K-chunk staged in LDS per iteration
#define NCHUNK   (KDIM / KC) // 36
#define ASTR     68          // m-major row stride: banks (4m+k)%64, conflict-free
#define NTHREADS 512         // 16 waves -> 16x16 N-tiles cover N=256

struct SampEntry {           // 32 bytes: channel-invariant bilinear record
    int   i00, i01, i10, i11;    // clamped linear indices into a 64x64 plane
    float w00, w01, w10, w11;    // bilinear weights, zeroed when out-of-bounds
};

__device__ __forceinline__ v8f wmma4(v2f a, v2f b, v8f c) {
    return __builtin_amdgcn_wmma_f32_16x16x4_f32(false, a, false, b,
                                                 (short)0, c, false, false);
}

// ---------------------------------------------------------------------------
// Kernel A: offset conv (18ch, 3x3, dil=2, pad=2) fused with bilinear
// coefficient construction. One thread per (b, pixel, tap).
// ---------------------------------------------------------------------------
__global__ __launch_bounds__(256) void build_sample_table(
    const float* __restrict__ x, const float* __restrict__ w_off,
    const float* __restrict__ b_off, SampEntry* __restrict__ tab)
{
    int tid = blockIdx.x * blockDim.x + threadIdx.x;   // exactly B_*PIX*K2 threads
    int tap = tid % K2;
    int bp  = tid / K2;
    int pix = bp % PIX;
    int b   = bp / PIX;
    int ho  = pix >> 6;
    int wo  = pix & 63;

    int co_y = 2 * tap, co_x = 2 * tap + 1;
    float dy = b_off[co_y];
    float dx = b_off[co_x];
    const float* xb = x + (size_t)b * C_ * PIX;
    const float* wy = w_off + (size_t)co_y * KDIM;     // w_off[co][c][ky][kx]
    const float* wx = w_off + (size_t)co_x * KDIM;

    #pragma unroll
    for (int ky = 0; ky < 3; ++ky) {
        int iy = ho - PAD_ + ky * DIL_;
        if (iy < 0 || iy >= H_) continue;
        #pragma unroll
        for (int kx = 0; kx < 3; ++kx) {
            int ix = wo - PAD_ + kx * DIL_;
            if (ix < 0 || ix >= W_) continue;
            const float* xp = xb + iy * W_ + ix;
            int wi = ky * 3 + kx;
            float ay = 0.f, ax = 0.f;
            for (int c = 0; c < C_; ++c) {
                float v = xp[(size_t)c * PIX];
                ay = fmaf(v, wy[c * K2 + wi], ay);
                ax = fmaf(v, wx[c * K2 + wi], ax);
            }
            dy += ay; dx += ax;
        }
    }

    float py = (float)(ho - PAD_ + (tap / 3) * DIL_) + dy;
    float px = (float)(wo - PAD_ + (tap % 3) * DIL_) + dx;
    float y0f = floorf(py), x0f = floorf(px);
    float ay = py - y0f, ax = px - x0f;
    int y0 = (int)y0f, x0 = (int)x0f;

    SampEntry e;
    auto mk = [&](int yy, int xx, float wgt, int& idx, float& wout) {
        bool valid = (yy >= 0) && (yy < H_) && (xx >= 0) && (xx < W_);
        int cy = min(max(yy, 0), H_ - 1);
        int cx = min(max(xx, 0), W_ - 1);
        idx  = cy * W_ + cx;
        wout = valid ? wgt : 0.0f;
    };
    mk(y0,     x0,     (1.f - ay) * (1.f - ax), e.i00, e.w00);
    mk(y0,     x0 + 1, (1.f - ay) * ax,         e.i01, e.w01);
    mk(y0 + 1, x0,     ay * (1.f - ax),         e.i10, e.w10);
    mk(y0 + 1, x0 + 1, ay * ax,                 e.i11, e.w11);
    tab[tid] = e;
}

// ---------------------------------------------------------------------------
// Kernel B: fused sample + GEMM. One workgroup = 64 pixels x all 256 output
// channels. 16 waves; each wave owns four 16x16 accumulators sharing one B
// fragment per K-step. B fragments of the current chunk are pre-issued into
// registers BEFORE the next-chunk fill gathers, so (in-order) LOADcnt waits
// on B do not drag the gathers in. Double-buffered LDS A, 1 barrier/chunk.
// ---------------------------------------------------------------------------
__global__ __launch_bounds__(NTHREADS) void deform_gemm(
    const float* __restrict__ x, const float* __restrict__ w,
    const SampEntry* __restrict__ tab, float* __restrict__ out)
{
    __shared__ SampEntry ltab[TM * K2];     // 576 records, 18432 B
    __shared__ float     As[2][TM * ASTR];  // m-major, stride 68: 34816 B

    int t    = threadIdx.x;
    int m0   = blockIdx.x * TM;
    int b    = m0 / PIX;
    int pix0 = m0 % PIX;                    // one full output row, same batch

    // --- stage sampling records via ASYNC copy (18432 B = 1152 x b128) -----
    {
        const char* src = (const char*)(tab + ((size_t)b * PIX + pix0) * K2);
        uint32_t dstb = (uint32_t)(uintptr_t)(void*)ltab;  // LDS byte offset
        for (int i = t; i < (TM * K2 * 32) / 16; i += NTHREADS) {
            uint32_t ld = dstb + i * 16;
            const char* gp = src + (size_t)i * 16;
            asm volatile("global_load_async_to_lds_b128 %0, %1, off"
                         :: "v"(ld), "v"(gp) : "memory");
        }
        asm volatile("s_wait_asynccnt 0x0" ::: "memory");
    }

    int lane  = t & 31;
    int wid   = t >> 5;                     // 0..15 -> N-block
    int n0    = wid * 16;
    int nlane = lane & 15;                  // M (A-frag) / N (B-frag) lane index
    int koff  = (lane >> 4) * 2;            // lanes 16-31 hold K=2,3
    const float* wrow = w + (size_t)(n0 + nlane) * KDIM;
    const float* xb   = x + (size_t)b * C_ * PIX;

    v8f acc[4];
    #pragma unroll
    for (int s = 0; s < 4; ++s) acc[s] = (v8f){0.f,0.f,0.f,0.f,0.f,0.f,0.f,0.f};

    // fill: thread t samples 8 consecutive k for pixel row m -> 2x ds_store_b128
    auto fill = [&](int p, int kk) {
        int m  = t >> 3;                    // 0..63
        int k8 = (t & 7) << 3;              // 0,8,...,56
        float v[8];
        #pragma unroll
        for (int j = 0; j < 8; ++j) {
            int kidx = kk + k8 + j;
            int c    = kidx / 9;
            int tap  = kidx - 9 * c;
            SampEntry en = ltab[m * K2 + tap];
            const float* xp = xb + (size_t)c * PIX;
            v[j] = en.w00 * xp[en.i00] + en.w01 * xp[en.i01]
                 + en.w10 * xp[en.i10] + en.w11 * xp[en.i11];
        }
        *(v4f*)&As[p][m * ASTR + k8]     = (v4f){v[0], v[1], v[2], v[3]};
        *(v4f*)&As[p][m * ASTR + k8 + 4] = (v4f){v[4], v[5], v[6], v[7]};
    };

    fill(0, 0);
    __syncthreads();
    for (int ck = 0; ck < NCHUNK; ++ck) {
        int kk = ck * KC;
        int p  = ck & 1;

        // pre-issue this chunk's 16 B fragments (one b64 clause, from L2)
        v2f breg[KC / 4];
        #pragma unroll
        for (int kb = 0; kb < KC; kb += 4)
            breg[kb >> 2] = *(const v2f*)(wrow + kk + kb + koff);

        if (ck + 1 < NCHUNK) {
            __builtin_prefetch(wrow + kk + KC, 0, 1);        // global_prefetch_b8
            fill(p ^ 1, kk + KC);                            // gathers drain under WMMA
        }

        const float* abase = &As[p][nlane * ASTR];
        #pragma unroll
        for (int kb = 0; kb < KC; kb += 4) {
            v2f bf = breg[kb >> 2];
            #pragma unroll
            for (int s = 0; s < 4; ++s) {
                v2f a = *(const v2f*)(abase + s * (16 * ASTR) + kb + koff);
                acc[s] = wmma4(a, bf, acc[s]);               // 4 WMMA share bf
            }
        }
        __syncthreads();
    }

    // D layout: VGPR i -> M = i + 8*(lane>=16); N = n0 + (lane&15)
    int n     = n0 + nlane;
    int rbase = (lane >> 4) * 8;
    float* op = out + ((size_t)b * O_ + n) * PIX + pix0;
    #pragma unroll
    for (int s = 0; s < 4; ++s)
        #pragma unroll
        for (int i = 0; i < 8; ++i) op[16 * s + rbase + i] = acc[s][i];
}

extern "C" void kernel_launch(void* const* d_in, const int* in_sizes, int n_in,
                              void* d_out, int out_size, void* d_ws, size_t ws_size,
                              hipStream_t stream)
{
    (void)in_sizes; (void)n_in; (void)out_size; (void)ws_size;
    const float* x     = (const float*)d_in[0];
    const float* w_off = (const float*)d_in[1];
    const float* b_off = (const float*)d_in[2];
    const float* w     = (const float*)d_in[3];
    float* out         = (float*)d_out;
    SampEntry* tab     = (SampEntry*)d_ws;   // 4*4096*9*32B = 4.7 MB

    // Kernel A: one thread per (b, pixel, tap) = 147456 threads
    build_sample_table<<<(B_ * PIX * K2) / 256, 256, 0, stream>>>(x, w_off, b_off, tab);
    // Kernel B: 256 workgroups of 512 threads (16 waves)
    deform_gemm<<<(B_ * PIX) / TM, NTHREADS, 0, stream>>>(x, w, tab, out);
}